// ALIGNNFF2_15444702396709
// MI455X (gfx1250) — compile-verified
//
#include <hip/hip_runtime.h>
#include <math.h>

#define HD   256
#define NNODE 4000
#define NEDGE 48000
#define NLG  120000
#define NGRAPH 50

typedef __attribute__((ext_vector_type(2))) float v2f;
typedef __attribute__((ext_vector_type(8))) float v8f;

// ---------------------------------------------------------------------------
// GEMM: Out[N,Dout] = Ain[N,Din] @ W[Din,Dout] + bias[Dout]
// Block = 128 threads = 4 wave32; each wave computes one 16x16 tile using
// V_WMMA_F32_16X16X4_F32. A-tile (16 x Din) staged in LDS, shared by waves.
// Requires: N%16==0, Din%4==0, Dout%64==0 (grid.y=Dout/64).
// ---------------------------------------------------------------------------
__global__ __launch_bounds__(128) void gemm_bias_wmma(
    const float* __restrict__ Ain, const float* __restrict__ W,
    const float* __restrict__ bias, float* __restrict__ Out,
    int N, int Din, int Dout)
{
  __shared__ float Ash[16 * 256];
  const int row0 = blockIdx.x << 4;
  const int wave = threadIdx.x >> 5;
  const int lane = threadIdx.x & 31;
  const int ncol = (((blockIdx.y << 2) + wave) << 4) + (lane & 15);

  // cooperative load of A tile [16 x Din] (row-major in LDS)
  const int nvec = Din << 2;  // 16*Din/4
  for (int v = threadIdx.x; v < nvec; v += 128) {
    const int elem = v << 2;
    const int r = elem / Din;
    const int k = elem - r * Din;
    *(float4*)(Ash + elem) =
        *(const float4*)(Ain + (size_t)(row0 + r) * Din + k);
  }
  __syncthreads();

  const int m  = lane & 15;
  const int kb = (lane >> 4) << 1;       // 0 for lanes 0-15, 2 for lanes 16-31
  const float bval = bias[ncol];

  v8f acc;
#pragma unroll
  for (int j = 0; j < 8; ++j) acc[j] = bval;

  const float* __restrict__ Wc = W + ncol;
  const float* __restrict__ Am = Ash + m * Din + kb;
  for (int k = 0; k < Din; k += 4) {
    v2f a, b;
    a.x = Am[k];
    a.y = Am[k + 1];
    b.x = Wc[(k + kb) * Dout];
    b.y = Wc[(k + kb + 1) * Dout];
    acc = __builtin_amdgcn_wmma_f32_16x16x4_f32(
        /*neg_a=*/false, a, /*neg_b=*/false, b,
        /*c_mod=*/(short)0, acc, /*reuse_a=*/false, /*reuse_b=*/false);
  }

  const int rbase = row0 + ((lane >> 4) << 3);
#pragma unroll
  for (int j = 0; j < 8; ++j)
    Out[(size_t)(rbase + j) * Dout + ncol] = acc[j];
}

// ---------------------------------------------------------------------------
// LayerNorm + SiLU. One wave32 per row (H = 64 or 256).
// mode 0: out = silu(ln(T))
// mode 1: out += silu(ln(T))
// mode 2: out += silu(ln(T + NUM/(DEN+1e-6)))
// ---------------------------------------------------------------------------
__global__ __launch_bounds__(256) void ln_silu_kernel(
    const float* __restrict__ T, const float* __restrict__ NUM,
    const float* __restrict__ DEN, const float* __restrict__ g,
    const float* __restrict__ bl, float* __restrict__ out,
    int N, int H, int mode)
{
  const int row = (blockIdx.x << 3) + (threadIdx.x >> 5);
  if (row >= N) return;
  const int lane = threadIdx.x & 31;
  const int per = H >> 5;  // 2 or 8
  const float* __restrict__ Trow = T + (size_t)row * H;
  float vals[8];
  float s = 0.f, ss = 0.f;
  for (int i = 0; i < per; ++i) {
    const int c = (i << 5) + lane;
    float v = Trow[c];
    if (mode == 2) {
      const size_t idx = (size_t)row * H + c;
      v += NUM[idx] / (DEN[idx] + 1e-6f);
    }
    vals[i] = v;
    s += v;
    ss += v * v;
  }
  for (int off = 16; off; off >>= 1) {
    s  += __shfl_xor(s, off, 32);
    ss += __shfl_xor(ss, off, 32);
  }
  const float mu  = s / (float)H;
  const float var = ss / (float)H - mu * mu;
  const float inv = rsqrtf(var + 1e-5f);
  float* __restrict__ Orow = out + (size_t)row * H;
  for (int i = 0; i < per; ++i) {
    const int c = (i << 5) + lane;
    const float t  = (vals[i] - mu) * inv * g[c] + bl[c];
    const float sl = t / (1.f + expf(-t));
    if (mode == 0) Orow[c] = sl; else Orow[c] += sl;
  }
}

// Fused gate: m = M + e_src[src] + e_dst[dst]; M = m; sigma = sigmoid(m);
// NUM[dst] += bh[src]*sigma; DEN[dst] += sigma.  (NUM/DEN distinct from A/B)
__global__ void gate_fused_kernel(float* __restrict__ M,
                                  const float* __restrict__ A,
                                  const float* __restrict__ B,
                                  const float* __restrict__ C,
                                  float* __restrict__ NUM,
                                  float* __restrict__ DEN,
                                  const int* __restrict__ src,
                                  const int* __restrict__ dst, int Ne)
{
  const int idx = blockIdx.x * 256 + threadIdx.x;
  if (idx >= Ne * HD) return;
  const int e = idx >> 8, f = idx & 255;
  const int sI = src[e], dI = dst[e];
  const float m = M[idx] + A[((size_t)sI << 8) + f] + B[((size_t)dI << 8) + f];
  M[idx] = m;
  const float sg = 1.f / (1.f + expf(-m));
  const size_t d = ((size_t)dI << 8) + f;
  atomicAdd(NUM + d, C[((size_t)sI << 8) + f] * sg);
  atomicAdd(DEN + d, sg);
}

// Fallback two-pass versions (NUM/DEN alias A/B):
__global__ void gather_add_kernel(float* __restrict__ M,
                                  const float* __restrict__ A,
                                  const float* __restrict__ B,
                                  const int* __restrict__ src,
                                  const int* __restrict__ dst, int Ne)
{
  const int idx = blockIdx.x * 256 + threadIdx.x;
  if (idx >= Ne * HD) return;
  const int e = idx >> 8, f = idx & 255;
  M[idx] += A[((size_t)src[e] << 8) + f] + B[((size_t)dst[e] << 8) + f];
}

__global__ void scatter_kernel(const float* __restrict__ M,
                               const float* __restrict__ C,
                               float* __restrict__ NUM, float* __restrict__ DEN,
                               const int* __restrict__ src,
                               const int* __restrict__ dst, int Ne)
{
  const int idx = blockIdx.x * 256 + threadIdx.x;
  if (idx >= Ne * HD) return;
  const int e = idx >> 8, f = idx & 255;
  const float sg = 1.f / (1.f + expf(-M[idx]));
  const size_t d = ((size_t)dst[e] << 8) + f;
  atomicAdd(NUM + d, C[((size_t)src[e] << 8) + f] * sg);
  atomicAdd(DEN + d, sg);
}

// RBF of cutoff-mapped bond length, 80 bins on [0,8]
__global__ void rbf_edge_kernel(const float* __restrict__ r,
                                float* __restrict__ out, int Ne)
{
  const int idx = blockIdx.x * 256 + threadIdx.x;
  if (idx >= Ne * 80) return;
  const int e = idx / 80, bin = idx - e * 80;
  const float x0 = r[e * 3], x1 = r[e * 3 + 1], x2 = r[e * 3 + 2];
  const float d = sqrtf(x0 * x0 + x1 * x1 + x2 * x2);
  const float xr = d * 0.25f;                       // d / rc, rc=4
  const float x3 = xr * xr * xr;                    // p=3 polynomial cutoff
  float c = 1.f - 10.f * x3 + 15.f * x3 * xr - 6.f * x3 * xr * xr;
  const float bl = (d <= 4.f) ? c : 0.f;
  const float center = (8.f / 79.f) * (float)bin;
  const float t = bl - center;
  out[idx] = expf(-(79.f / 8.f) * t * t);
}

// RBF of bond cosine, 40 bins on [-1,1]
__global__ void rbf_angle_kernel(const float* __restrict__ r,
                                 const int* __restrict__ lg_src,
                                 const int* __restrict__ lg_dst,
                                 float* __restrict__ out, int Nl)
{
  const int idx = blockIdx.x * 256 + threadIdx.x;
  if (idx >= Nl * 40) return;
  const int e = idx / 40, bin = idx - e * 40;
  const int a = lg_src[e], b = lg_dst[e];
  const float r1x = -r[a * 3], r1y = -r[a * 3 + 1], r1z = -r[a * 3 + 2];
  const float r2x =  r[b * 3], r2y =  r[b * 3 + 1], r2z =  r[b * 3 + 2];
  const float dp = r1x * r2x + r1y * r2y + r1z * r2z;
  const float n1 = sqrtf(r1x * r1x + r1y * r1y + r1z * r1z);
  const float n2 = sqrtf(r2x * r2x + r2y * r2y + r2z * r2z);
  float cosv = dp / (n1 * n2);
  cosv = fminf(1.f, fmaxf(-1.f, cosv));
  const float center = -1.f + (2.f / 39.f) * (float)bin;
  const float t = cosv - center;
  out[idx] = expf(-(39.f / 2.f) * t * t);
}

__global__ void pool_scatter_kernel(const float* __restrict__ x,
                                    const int* __restrict__ gid,
                                    float* __restrict__ pool,
                                    float* __restrict__ cnt, int Nn)
{
  const int idx = blockIdx.x * 256 + threadIdx.x;
  if (idx >= Nn * HD) return;
  const int n = idx >> 8, f = idx & 255;
  const int g = gid[n];
  atomicAdd(pool + ((size_t)g << 8) + f, x[idx]);
  if (f == 0) atomicAdd(cnt + g, 1.f);
}

__global__ void pool_final_kernel(const float* __restrict__ pool,
                                  const float* __restrict__ cnt,
                                  const float* __restrict__ fcw,
                                  const float* __restrict__ fcb,
                                  float* __restrict__ out)
{
  __shared__ float red[256];
  const int g = blockIdx.x, f = threadIdx.x;
  const float c = fmaxf(cnt[g], 1.f);
  red[f] = (pool[((size_t)g << 8) + f] / c) * fcw[f];
  __syncthreads();
  for (int s = 128; s; s >>= 1) {
    if (f < s) red[f] += red[f + s];
    __syncthreads();
  }
  if (f == 0) out[g] = red[0] + fcb[0];
}

// ---------------------------------------------------------------------------
static inline void gemm(const float* A, const float* W, const float* b,
                        float* O, int N, int Din, int Dout, hipStream_t s)
{
  dim3 grid(N / 16, Dout / 64);
  hipLaunchKernelGGL(gemm_bias_wmma, grid, dim3(128), 0, s, A, W, b, O, N, Din, Dout);
}

static inline void ln_silu(const float* T, const float* NUM, const float* DEN,
                           const float* g, const float* bl, float* out,
                           int N, int H, int mode, hipStream_t s)
{
  hipLaunchKernelGGL(ln_silu_kernel, dim3((N + 7) / 8), dim3(256), 0, s,
                     T, NUM, DEN, g, bl, out, N, H, mode);
}

// EdgeGatedGraphConv. Param leaf order (insertion order):
// 0 src_gate.w 1 .b | 2 dst_gate.w 3 .b | 4 edge_gate.w 5 .b
// 6 src_update.w 7 .b | 8 dst_update.w 9 .b | 10 lnN.g 11 lnN.b | 12 lnE.g 13 lnE.b
static void run_eggc(const float* const* P, float* Xn, float* Ye,
                     const int* src, const int* dst, int Nn, int Ne,
                     float* A, float* B, float* C, float* M,
                     float* NUM, float* DEN, bool fused, hipStream_t s)
{
  gemm(Xn, P[0], P[1], A, Nn, HD, HD, s);                 // e_src
  gemm(Xn, P[2], P[3], B, Nn, HD, HD, s);                 // e_dst
  gemm(Xn, P[8], P[9], C, Nn, HD, HD, s);                 // bh (dst_update)
  gemm(Ye, P[4], P[5], M, Ne, HD, HD, s);                 // edge_gate(y)
  const int nblk = (Ne * HD + 255) / 256;
  if (fused) {
    hipMemsetAsync(NUM, 0, (size_t)Nn * HD * sizeof(float), s);
    hipMemsetAsync(DEN, 0, (size_t)Nn * HD * sizeof(float), s);
    hipLaunchKernelGGL(gate_fused_kernel, dim3(nblk), dim3(256), 0, s,
                       M, A, B, C, NUM, DEN, src, dst, Ne);
  } else {
    // NUM aliases A, DEN aliases B: needs the two-pass schedule
    hipLaunchKernelGGL(gather_add_kernel, dim3(nblk), dim3(256), 0, s,
                       M, A, B, src, dst, Ne);
    hipMemsetAsync(NUM, 0, (size_t)Nn * HD * sizeof(float), s);
    hipMemsetAsync(DEN, 0, (size_t)Nn * HD * sizeof(float), s);
    hipLaunchKernelGGL(scatter_kernel, dim3(nblk), dim3(256), 0, s,
                       M, C, NUM, DEN, src, dst, Ne);
  }
  gemm(Xn, P[6], P[7], C, Nn, HD, HD, s);                 // src_update (alias C)
  ln_silu(C, NUM, DEN, P[10], P[11], Xn, Nn, HD, 2, s);   // x += silu(ln(xu+h))
  ln_silu(M, nullptr, nullptr, P[12], P[13], Ye, Ne, HD, 1, s); // y += silu(ln(m))
}

extern "C" void kernel_launch(void* const* d_in, const int* in_sizes, int n_in,
                              void* d_out, int out_size, void* d_ws, size_t ws_size,
                              hipStream_t stream)
{
  (void)in_sizes; (void)n_in; (void)out_size;
  const float* atomf  = (const float*)d_in[0];
  const float* r      = (const float*)d_in[1];
  const int*   src    = (const int*)d_in[2];
  const int*   dst    = (const int*)d_in[3];
  const int*   lg_src = (const int*)d_in[4];
  const int*   lg_dst = (const int*)d_in[5];
  const int*   gid    = (const int*)d_in[6];
  // params: recursive insertion-order flatten -> 190 float* leaves
  const float* const* P = (const float* const*)(d_in + 7);

  float* ws = (float*)d_ws;
  size_t o = 0;
  float* X = ws + o; o += (size_t)NNODE * HD;
  float* Y = ws + o; o += (size_t)NEDGE * HD;
  float* Z = ws + o; o += (size_t)NLG * HD;
  float* A = ws + o; o += (size_t)NEDGE * HD;   // e_src / rbf scratch
  float* B = ws + o; o += (size_t)NEDGE * HD;   // e_dst / emb1 scratch
  float* C = ws + o; o += (size_t)NEDGE * HD;   // bh / xu
  float* M = ws + o; o += (size_t)NLG * HD;     // gate pre-activations
  float* POOL = ws + o; o += (size_t)NGRAPH * HD;
  float* CNT  = ws + o; o += 64;
  // optional dedicated segment-sum buffers (fused single-pass gate stage)
  const size_t need_fused = (o + 2 * (size_t)NEDGE * HD) * sizeof(float);
  const bool fused = ws_size >= need_fused;
  float* NUM = fused ? (ws + o) : A;  o += fused ? (size_t)NEDGE * HD : 0;
  float* DEN = fused ? (ws + o) : B;

  // ---- edge embedding: rbf(cutoff(|r|)) -> 64 -> 256 -> Y
  hipLaunchKernelGGL(rbf_edge_kernel, dim3((NEDGE * 80 + 255) / 256), dim3(256),
                     0, stream, r, A, NEDGE);
  gemm(A, P[4], P[5], B, NEDGE, 80, 64, stream);
  ln_silu(B, nullptr, nullptr, P[6], P[7], B, NEDGE, 64, 0, stream);
  gemm(B, P[8], P[9], Y, NEDGE, 64, HD, stream);
  ln_silu(Y, nullptr, nullptr, P[10], P[11], Y, NEDGE, HD, 0, stream);

  // ---- angle embedding: rbf(cos) -> 64 -> 256 -> Z
  hipLaunchKernelGGL(rbf_angle_kernel, dim3((NLG * 40 + 255) / 256), dim3(256),
                     0, stream, r, lg_src, lg_dst, A, NLG);
  gemm(A, P[12], P[13], B, NLG, 40, 64, stream);
  ln_silu(B, nullptr, nullptr, P[14], P[15], B, NLG, 64, 0, stream);
  gemm(B, P[16], P[17], Z, NLG, 64, HD, stream);
  ln_silu(Z, nullptr, nullptr, P[18], P[19], Z, NLG, HD, 0, stream);

  // ---- atom embedding -> X
  gemm(atomf, P[0], P[1], X, NNODE, 92, HD, stream);
  ln_silu(X, nullptr, nullptr, P[2], P[3], X, NNODE, HD, 0, stream);

  // ---- 4 ALIGNN layers (node conv + line-graph conv)
  for (int l = 0; l < 4; ++l) {
    run_eggc(P + 20 + l * 28,      X, Y, src, dst,       NNODE, NEDGE,
             A, B, C, M, NUM, DEN, fused, stream);
    run_eggc(P + 20 + l * 28 + 14, Y, Z, lg_src, lg_dst, NEDGE, NLG,
             A, B, C, M, NUM, DEN, fused, stream);
  }
  // ---- 4 GCN layers
  for (int l = 0; l < 4; ++l)
    run_eggc(P + 132 + l * 14, X, Y, src, dst, NNODE, NEDGE,
             A, B, C, M, NUM, DEN, fused, stream);

  // ---- average pooling + final linear
  hipMemsetAsync(POOL, 0, (size_t)NGRAPH * HD * sizeof(float), stream);
  hipMemsetAsync(CNT, 0, 64 * sizeof(float), stream);
  hipLaunchKernelGGL(pool_scatter_kernel, dim3((NNODE * HD + 255) / 256),
                     dim3(256), 0, stream, X, gid, POOL, CNT, NNODE);
  hipLaunchKernelGGL(pool_final_kernel, dim3(NGRAPH), dim3(256), 0, stream,
                     POOL, CNT, P[188], P[189], (float*)d_out);
}